// WinnerTakesAll_32908039422109
// MI455X (gfx1250) — compile-verified
//
#include <hip/hip_runtime.h>
#include <stdint.h>

// Problem constants (from reference): x is (32, 32, 256, 256) f32.
#define NBATCH 32
#define NPB (32 * 256 * 256)   // 2,097,152 elements per batch
#define TOPK 64
#define NBINS 4096             // histogram over top 12 bits of order-key
#define CAND_MAX 6144          // per-batch candidate cap (expected ~3K)

// pass-1 tiling
#define CHUNK 65536            // elements per block (32 blocks per batch)
#define TILE  4096             // elements staged in LDS per step (16 KB)
#define TILES (CHUNK / TILE)   // 16

typedef float floatx4 __attribute__((ext_vector_type(4)));

__device__ __forceinline__ unsigned orderKey(float f) {
  unsigned b = __float_as_uint(f);
  return (b & 0x80000000u) ? ~b : (b | 0x80000000u);
}

__device__ __forceinline__ float keyToFloat(unsigned k) {
  unsigned b = (k & 0x80000000u) ? (k ^ 0x80000000u) : ~k;
  return __uint_as_float(b);
}

// ---------------------------------------------------------------------------
// Kernel 0: zero the per-batch candidate counters (deterministic every call).
// ---------------------------------------------------------------------------
__global__ __launch_bounds__(64) void k_zero(unsigned* p) {
  p[threadIdx.x] = 0u;
}

// ---------------------------------------------------------------------------
// Kernel 1 (fused streaming pass), grid = (NPB/CHUNK, NBATCH), block = 256:
//   Phase A: double-buffered global->LDS async copy of the 64K-element chunk;
//            block-local 4096-bin histogram of key>>20 via LDS atomics;
//            simultaneous non-temporal zero-fill of the output chunk
//            (lane-major mapping: coalesced 512B wave bursts, conflict-free
//            LDS b128 reads).
//   Phase B: parallel suffix scan of the histogram -> block bucket floor of
//            the chunk's 64th-largest key.
//   Phase C: re-read own chunk (L2-hot), append candidates >= floor to the
//            per-batch candidate buffer (packed u64: key<<32 | ~idx).
// ---------------------------------------------------------------------------
__global__ __launch_bounds__(256) void k_pass1(
    const float* __restrict__ x, float* __restrict__ out,
    unsigned* __restrict__ cand_cnt, unsigned long long* __restrict__ cand) {
  __shared__ float tilebuf[2][TILE];   // 32 KB double-buffered staging
  __shared__ unsigned shist[NBINS];    // 16 KB histogram
  __shared__ unsigned ssum[256];       // segment suffix sums
  __shared__ unsigned sfloor;

  const int tid = threadIdx.x;
  const int b = blockIdx.y;
  const unsigned chunk_elem0 = (unsigned)blockIdx.x * CHUNK;  // within batch
  const float* gbase = x + (size_t)b * NPB + chunk_elem0;     // uniform
  floatx4* o4 = reinterpret_cast<floatx4*>(out + (size_t)b * NPB + chunk_elem0);
  const floatx4* x4c = reinterpret_cast<const floatx4*>(gbase);

  for (int i = tid; i < NBINS; i += 256) shist[i] = 0u;

  const unsigned ldsbase = (unsigned)(size_t)(&tilebuf[0][0]);

  // ---- Phase A: async-staged streaming histogram + zero-fill ----
  // issue helper: 4 rounds x 256 lanes x 16B = one 16 KB tile
#define ISSUE_TILE(T, BUF)                                                    \
  do {                                                                        \
    unsigned lb = ldsbase + (unsigned)((BUF) * TILE * 4);                     \
    _Pragma("unroll") for (int r = 0; r < 4; ++r) {                           \
      unsigned goff = (unsigned)(((T) * TILE + r * 1024 + tid * 4) * 4);      \
      unsigned laddr = lb + (unsigned)((r * 1024 + tid * 4) * 4);             \
      asm volatile("global_load_async_to_lds_b128 %0, %1, %2"                 \
                   :                                                          \
                   : "v"(laddr), "v"(goff), "s"(gbase)                        \
                   : "memory");                                               \
    }                                                                         \
  } while (0)

  ISSUE_TILE(0, 0);
  for (int t = 0; t < TILES; ++t) {
    if (t + 1 < TILES) {
      ISSUE_TILE(t + 1, (t + 1) & 1);
      // in-order completion: <=4 outstanding means tile t has landed
      asm volatile("s_wait_asynccnt 0x4" ::: "memory");
    } else {
      asm volatile("s_wait_asynccnt 0x0" ::: "memory");
    }
    __syncthreads();  // all waves' async data visible

    const floatx4* tb = reinterpret_cast<const floatx4*>(&tilebuf[t & 1][0]);
    const floatx4 z4 = {0.f, 0.f, 0.f, 0.f};
#pragma unroll
    for (int e = 0; e < 4; ++e) {
      // lane-major: lanes hit consecutive 16B -> coalesced global bursts,
      // optimal 2-phase LDS b128 access
      floatx4 v = tb[e * 256 + tid];
      __builtin_nontemporal_store(z4, &o4[t * (TILE / 4) + e * 256 + tid]);
      atomicAdd(&shist[orderKey(v.x) >> 20], 1u);
      atomicAdd(&shist[orderKey(v.y) >> 20], 1u);
      atomicAdd(&shist[orderKey(v.z) >> 20], 1u);
      atomicAdd(&shist[orderKey(v.w) >> 20], 1u);
    }
    __syncthreads();  // readers done before buffer is overwritten
  }
#undef ISSUE_TILE

  // ---- Phase B: find bucket of the block's 64th-largest key ----
  {
    unsigned s = 0;
#pragma unroll
    for (int i = 0; i < NBINS / 256; ++i) s += shist[tid * (NBINS / 256) + i];
    ssum[tid] = s;
  }
  __syncthreads();
  if (tid == 0) {
    unsigned acc = 0;
    int bucket = 0;
    for (int seg = 255; seg >= 0; --seg) {
      if (acc + ssum[seg] >= TOPK) {
        int lo = seg * (NBINS / 256);
        for (int bin = lo + (NBINS / 256) - 1; bin >= lo; --bin) {
          acc += shist[bin];
          if (acc >= TOPK) { bucket = bin; break; }
        }
        break;
      }
      acc += ssum[seg];
    }
    sfloor = ((unsigned)bucket) << 20;
  }
  __syncthreads();
  const unsigned floorkey = sfloor;

  // ---- Phase C: re-read own chunk (L2-hot) and append candidates ----
  for (int r = 0; r < CHUNK / 4 / 256; ++r) {
    int i4 = r * 256 + tid;
    __builtin_prefetch(&x4c[i4 + 256], 0, 2);
    floatx4 v = x4c[i4];
    float vv[4] = {v.x, v.y, v.z, v.w};
#pragma unroll
    for (int c = 0; c < 4; ++c) {
      unsigned key = orderKey(vv[c]);
      if (key >= floorkey) {
        unsigned idx = chunk_elem0 + (unsigned)(i4 * 4 + c);
        unsigned pos = atomicAdd(&cand_cnt[b], 1u);
        if (pos < CAND_MAX) {
          cand[(size_t)b * CAND_MAX + pos] =
              ((unsigned long long)key << 32) | (unsigned)(~idx);
        }
      }
    }
  }
}

// ---------------------------------------------------------------------------
// Kernel 2: per batch, exact top-K among candidates by rank counting on the
// packed u64 (value desc, index asc => matches jax top_k tie-breaking);
// scatter the K winners into the already-zeroed output.  grid = NBATCH.
// ---------------------------------------------------------------------------
__global__ __launch_bounds__(256) void k_select(
    float* __restrict__ out, const unsigned* __restrict__ cand_cnt,
    const unsigned long long* __restrict__ cand) {
  __shared__ unsigned long long sp[CAND_MAX];  // 48 KB

  const int tid = threadIdx.x;
  const int b = blockIdx.x;
  unsigned cnt = cand_cnt[b];
  if (cnt > CAND_MAX) cnt = CAND_MAX;

  for (unsigned i = tid; i < cnt; i += 256)
    sp[i] = cand[(size_t)b * CAND_MAX + i];
  __syncthreads();

  for (unsigned i = tid; i < cnt; i += 256) {
    const unsigned long long p = sp[i];
    int rank = 0;
    for (unsigned j = 0; j < cnt; ++j) {
      rank += (sp[j] > p);
      if (rank >= TOPK) break;
    }
    if (rank < TOPK) {
      unsigned idx = ~(unsigned)p;
      unsigned key = (unsigned)(p >> 32);
      out[(size_t)b * NPB + idx] = keyToFloat(key);
    }
  }
}

// ---------------------------------------------------------------------------
// Host-side launcher.
// ---------------------------------------------------------------------------
extern "C" void kernel_launch(void* const* d_in, const int* in_sizes, int n_in,
                              void* d_out, int out_size, void* d_ws,
                              size_t ws_size, hipStream_t stream) {
  const float* x = (const float*)d_in[0];
  float* out = (float*)d_out;
  unsigned char* ws = (unsigned char*)d_ws;

  // scratch layout (~1.6 MB total)
  unsigned* cand_cnt = (unsigned*)(ws + 0);                       // 256 B used
  unsigned long long* cand = (unsigned long long*)(ws + 1024);    // 32*6144*8

  // 0) zero candidate counters
  k_zero<<<1, 64, 0, stream>>>(cand_cnt);

  // 1) fused streaming pass: histogram + zero-fill + candidate collection
  {
    dim3 grid(NPB / CHUNK, NBATCH);  // (32, 32)
    k_pass1<<<grid, 256, 0, stream>>>(x, out, cand_cnt, cand);
  }

  // 2) exact per-batch top-K, scatter winners
  k_select<<<NBATCH, 256, 0, stream>>>(out, cand_cnt, cand);
}